// HomoGNN_88768384074106
// MI455X (gfx1250) — compile-verified
//
#include <hip/hip_runtime.h>
#include <math.h>

// ---------------------------------------------------------------------------
// Problem constants (match reference)
// ---------------------------------------------------------------------------
#define D_FEAT 128
#define NEG_SLOPE 0.2f
#define BN_EPS 1e-12f

typedef __attribute__((ext_vector_type(2))) float v2f;
typedef __attribute__((ext_vector_type(8))) float v8f;

// Monotonic encoding of float -> uint so atomicMax(u32) == float max.
__device__ __forceinline__ unsigned encf(float f) {
  unsigned u = __float_as_uint(f);
  return (u & 0x80000000u) ? ~u : (u | 0x80000000u);
}
__device__ __forceinline__ float decf(unsigned u) {
  unsigned v = (u & 0x80000000u) ? (u & 0x7FFFFFFFu) : ~u;
  return __uint_as_float(v);
}
__device__ __forceinline__ float leaky(float e) {
  return e > 0.0f ? e : NEG_SLOPE * e;
}

// ---------------------------------------------------------------------------
// K1: h = x @ W   (fp32 WMMA, 16x16 tile per wave, K stepped by 4)
// A 16x4 f32: lanes 0-15 hold M=0-15 {K=k,k+1}; lanes 16-31 hold {K=k+2,k+3}
// B 4x16 f32: lanes 0-15 hold N=0-15 {K=k,k+1}; lanes 16-31 hold {K=k+2,k+3}
// C/D 16x16 f32: vgpr v -> row (v + 8*half), col = lane&15
// ---------------------------------------------------------------------------
__global__ __launch_bounds__(256)
void gat_gemm_wmma_f32(const float* __restrict__ X, const float* __restrict__ W,
                       float* __restrict__ H, int M) {
  const int lane = threadIdx.x & 31;
  const int wave = threadIdx.x >> 5;      // 8 waves -> 8 column tiles (128 cols)
  const int mbase = blockIdx.x * 16;
  const int nbase = wave * 16;
  const int half = lane >> 4;             // 0: lanes 0-15, 1: lanes 16-31
  const int l15 = lane & 15;

  int arow = mbase + l15;
  if (arow >= M) arow = M - 1;            // clamp loads (stores guarded below)
  const float* __restrict__ xrow = X + (size_t)arow * D_FEAT;

  v8f c = {};
#pragma unroll 4
  for (int k = 0; k < D_FEAT; k += 4) {
    const int ka = k + 2 * half;
    v2f a, b;
    a.x = xrow[ka];
    a.y = xrow[ka + 1];
    b.x = W[(size_t)ka * D_FEAT + nbase + l15];
    b.y = W[(size_t)(ka + 1) * D_FEAT + nbase + l15];
    c = __builtin_amdgcn_wmma_f32_16x16x4_f32(
        /*neg_a=*/false, a, /*neg_b=*/false, b,
        /*c_mod=*/(short)0, c, /*reuse_a=*/false, /*reuse_b=*/false);
  }

  // Store epilogue: mbase/M are wave-uniform -> one scalar branch instead of
  // eight exec-mask toggles. Fast path covers every block when M % 16 == 0.
  float* __restrict__ hbase = H + (size_t)(mbase + 8 * half) * D_FEAT + nbase + l15;
  if (mbase + 16 <= M) {
#pragma unroll
    for (int v = 0; v < 8; ++v) {
      hbase[(size_t)v * D_FEAT] = c[v];
    }
  } else {
#pragma unroll
    for (int v = 0; v < 8; ++v) {
      if (mbase + 8 * half + v < M) hbase[(size_t)v * D_FEAT] = c[v];
    }
  }
}

// ---------------------------------------------------------------------------
// K2: per-node attention scores hs = h.a_src, hd = h.a_dst  (wave per node)
// Also seeds segment-max with the self-loop score.
// ---------------------------------------------------------------------------
__global__ __launch_bounds__(256)
void gat_node_scores(const float* __restrict__ H, const float* __restrict__ asrc,
                     const float* __restrict__ adst, float* __restrict__ hs,
                     float* __restrict__ hd, unsigned* __restrict__ menc, int N) {
  const int node = (int)((blockIdx.x * (unsigned)blockDim.x + threadIdx.x) >> 5);
  const int lane = threadIdx.x & 31;
  if (node >= N) return;

  const float4 hv = ((const float4*)(H + (size_t)node * D_FEAT))[lane];
  const float4 as = ((const float4*)asrc)[lane];
  const float4 ad = ((const float4*)adst)[lane];
  float ss = hv.x * as.x + hv.y * as.y + hv.z * as.z + hv.w * as.w;
  float sd = hv.x * ad.x + hv.y * ad.y + hv.z * ad.z + hv.w * ad.w;
#pragma unroll
  for (int off = 16; off > 0; off >>= 1) {
    ss += __shfl_xor(ss, off, 32);
    sd += __shfl_xor(sd, off, 32);
  }
  if (lane == 0) {
    hs[node] = ss;
    hd[node] = sd;
    menc[node] = encf(leaky(ss + sd));   // self-loop seeds the max
  }
}

// ---------------------------------------------------------------------------
// K3: edge pass 1 - leaky score, stash e, atomic segment max on dst
// ---------------------------------------------------------------------------
__global__ __launch_bounds__(256)
void gat_edge_max(const int* __restrict__ src, const int* __restrict__ dst,
                  const float* __restrict__ hs, const float* __restrict__ hd,
                  float* __restrict__ ebuf, unsigned* __restrict__ menc, int E) {
  const int i = blockIdx.x * blockDim.x + threadIdx.x;
  if (i >= E) return;
  const int s = src[i], d = dst[i];
  const float e = leaky(hs[s] + hd[d]);
  ebuf[i] = e;
  atomicMax(&menc[d], encf(e));
}

// ---------------------------------------------------------------------------
// K4: per-node - decode max, self-loop exp, seed softmax denominator
// ---------------------------------------------------------------------------
__global__ __launch_bounds__(256)
void gat_node_selfexp(const unsigned* __restrict__ menc,
                      const float* __restrict__ hs, const float* __restrict__ hd,
                      float* __restrict__ mf, float* __restrict__ sden,
                      float* __restrict__ selfee, int N) {
  const int i = blockIdx.x * blockDim.x + threadIdx.x;
  if (i >= N) return;
  const float m = decf(menc[i]);
  const float se = __expf(leaky(hs[i] + hd[i]) - m);
  mf[i] = m;
  selfee[i] = se;
  sden[i] = se;        // self loop seeds the denominator
}

// ---------------------------------------------------------------------------
// K5: zero the feature accumulator (float4 stores)
// ---------------------------------------------------------------------------
__global__ __launch_bounds__(256)
void gat_zero_f4(float4* __restrict__ p, int n4) {
  const int i = blockIdx.x * blockDim.x + threadIdx.x;
  if (i < n4) p[i] = make_float4(0.f, 0.f, 0.f, 0.f);
}

// ---------------------------------------------------------------------------
// K6: edge pass 2 - ee = exp(e - m[dst]), atomic add to denominator
// ---------------------------------------------------------------------------
__global__ __launch_bounds__(256)
void gat_edge_expsum(const int* __restrict__ dst, float* __restrict__ ebuf,
                     const float* __restrict__ mf, float* __restrict__ sden, int E) {
  const int i = blockIdx.x * blockDim.x + threadIdx.x;
  if (i >= E) return;
  const int d = dst[i];
  const float ee = __expf(ebuf[i] - mf[d]);
  ebuf[i] = ee;
  atomicAdd(&sden[d], ee);
}

// ---------------------------------------------------------------------------
// K7: heavy aggregation - wave per edge, float4 gather of h[src] (L2-resident),
//     4x global_atomic_add_f32 per lane into acc[dst].
// ---------------------------------------------------------------------------
__global__ __launch_bounds__(256)
void gat_edge_aggregate(const int* __restrict__ src, const int* __restrict__ dst,
                        const float* __restrict__ ebuf, const float* __restrict__ sden,
                        const float* __restrict__ H, float* __restrict__ acc, int E) {
  const int e = (int)((blockIdx.x * (unsigned)blockDim.x + threadIdx.x) >> 5);
  const int lane = threadIdx.x & 31;
  if (e >= E) return;
  const int s = src[e], d = dst[e];
  const float alpha = ebuf[e] / sden[d];
  const float4 hv = ((const float4*)(H + (size_t)s * D_FEAT))[lane];
  float* out = acc + (size_t)d * D_FEAT + lane * 4;
  atomicAdd(out + 0, alpha * hv.x);
  atomicAdd(out + 1, alpha * hv.y);
  atomicAdd(out + 2, alpha * hv.z);
  atomicAdd(out + 3, alpha * hv.w);
}

// ---------------------------------------------------------------------------
// K8: finalize - add self-loop term + bias, BatchNorm(eval) + ReLU. In-place OK.
// ---------------------------------------------------------------------------
__global__ __launch_bounds__(256)
void gat_finalize(const float* __restrict__ H, const float* __restrict__ acc,
                  const float* __restrict__ selfee, const float* __restrict__ sden,
                  const float* __restrict__ bias, const float* __restrict__ gamma,
                  const float* __restrict__ beta, const float* __restrict__ mean,
                  const float* __restrict__ var, float* __restrict__ out, int total) {
  const int idx = blockIdx.x * blockDim.x + threadIdx.x;
  if (idx >= total) return;
  const int i = idx >> 7;          // node
  const int j = idx & (D_FEAT - 1);
  const float alpha_self = selfee[i] / sden[i];
  float v = acc[idx] + alpha_self * H[idx] + bias[j];
  v = (v - mean[j]) * (gamma[j] * rsqrtf(var[j] + BN_EPS)) + beta[j];
  out[idx] = v > 0.0f ? v : 0.0f;
}

// ---------------------------------------------------------------------------
// Host-side orchestration
// ---------------------------------------------------------------------------
extern "C" void kernel_launch(void* const* d_in, const int* in_sizes, int n_in,
                              void* d_out, int out_size, void* d_ws, size_t ws_size,
                              hipStream_t stream) {
  (void)n_in; (void)out_size; (void)ws_size;

  const float* x        = (const float*)d_in[0];
  const int*   eidx     = (const int*)d_in[1];     // [2, E] (harness int32)
  const float* Ws       = (const float*)d_in[2];   // [L, D, D]
  const float* a_src    = (const float*)d_in[3];   // [L, D]
  const float* a_dst    = (const float*)d_in[4];
  const float* bias     = (const float*)d_in[5];
  const float* gamma    = (const float*)d_in[6];
  const float* beta     = (const float*)d_in[7];
  const float* run_mean = (const float*)d_in[8];
  const float* run_var  = (const float*)d_in[9];

  const int N = in_sizes[0] / D_FEAT;
  const int E = in_sizes[1] / 2;
  const int* src = eidx;
  const int* dst = eidx + E;

  // Workspace carve-up (~56 MB): h, accA (layer-0 accum -> layer-1 input),
  // per-edge e/ee, per-node scalars.
  float*    h      = (float*)d_ws;
  float*    accA   = h + (size_t)N * D_FEAT;
  float*    ebuf   = accA + (size_t)N * D_FEAT;
  float*    hs     = ebuf + E;
  float*    hd     = hs + N;
  unsigned* menc   = (unsigned*)(hd + N);
  float*    mf     = (float*)(menc + N);
  float*    sden   = mf + N;
  float*    selfee = sden + N;

  const int nodeBlocks  = (N + 255) / 256;
  const int edgeBlocks  = (E + 255) / 256;
  const int waveNodeBl  = (N + 7) / 8;    // 8 waves per 256-thread block
  const int waveEdgeBl  = (E + 7) / 8;
  const int gemmBlocks  = (N + 15) / 16;
  const int featBlocks  = (int)(((size_t)N * D_FEAT + 255) / 256);
  const int zero4Blocks = (int)(((size_t)N * (D_FEAT / 4) + 255) / 256);

  for (int l = 0; l < 2; ++l) {
    const float* xin = (l == 0) ? x : accA;
    float*       acc = (l == 0) ? accA : (float*)d_out;   // finalized in place
    const float* W   = Ws + (size_t)l * D_FEAT * D_FEAT;

    gat_gemm_wmma_f32<<<gemmBlocks, 256, 0, stream>>>(xin, W, h, N);
    gat_node_scores<<<waveNodeBl, 256, 0, stream>>>(
        h, a_src + l * D_FEAT, a_dst + l * D_FEAT, hs, hd, menc, N);
    gat_edge_max<<<edgeBlocks, 256, 0, stream>>>(src, dst, hs, hd, ebuf, menc, E);
    gat_node_selfexp<<<nodeBlocks, 256, 0, stream>>>(menc, hs, hd, mf, sden, selfee, N);
    gat_zero_f4<<<zero4Blocks, 256, 0, stream>>>((float4*)acc, N * (D_FEAT / 4));
    gat_edge_expsum<<<edgeBlocks, 256, 0, stream>>>(dst, ebuf, mf, sden, E);
    gat_edge_aggregate<<<waveEdgeBl, 256, 0, stream>>>(src, dst, ebuf, sden, h, acc, E);
    gat_finalize<<<featBlocks, 256, 0, stream>>>(
        h, acc, selfee, sden, bias + l * D_FEAT, gamma + l * D_FEAT,
        beta + l * D_FEAT, run_mean + l * D_FEAT, run_var + l * D_FEAT,
        acc, N * D_FEAT);
  }
}